// MPNN6_46909632807730
// MI455X (gfx1250) — compile-verified
//
#include <hip/hip_runtime.h>
#include <math.h>

typedef __attribute__((ext_vector_type(2))) float v2f;
typedef __attribute__((ext_vector_type(8))) float v8f;

__device__ __forceinline__ float sigm(float v) { return 1.0f / (1.0f + expf(-v)); }

// ---------------- elementwise helpers ----------------
__global__ void k_zero(float* p, int n) {
  int i = blockIdx.x * blockDim.x + threadIdx.x;
  if (i < n) p[i] = 0.0f;
}

__global__ void k_deg(const int* __restrict__ src, const float* __restrict__ ea,
                      float* deg, int E) {
  int e = blockIdx.x * blockDim.x + threadIdx.x;
  if (e < E) atomicAdd(&deg[src[e]], ea[e]);
}

__global__ void k_dis(float* deg, int n) {
  int i = blockIdx.x * blockDim.x + threadIdx.x;
  if (i < n) { float d = deg[i]; deg[i] = d > 0.0f ? rsqrtf(d) : 0.0f; }
}

__global__ void k_wnorm(const int* __restrict__ src, const int* __restrict__ dst,
                        const float* __restrict__ ea, const float* __restrict__ dis,
                        float* __restrict__ w, int E) {
  int e = blockIdx.x * blockDim.x + threadIdx.x;
  if (e < E) w[e] = dis[src[e]] * ea[e] * dis[dst[e]];
}

// Pack Wx[4,5,Cx,Co] + Wh[4,5,Ch,Co] -> WB[5Cx+5Ch, 4Co]; bias = bx+bh+bg
__global__ void k_pack(const float* __restrict__ Wx, const float* __restrict__ Wh,
                       const float* __restrict__ bx, const float* __restrict__ bh,
                       const float* __restrict__ bg,
                       float* __restrict__ WB, float* __restrict__ cb,
                       int Cx, int Ch, int Co) {
  int Nout = 4 * Co;
  int Ktot = 5 * (Cx + Ch);
  int idx = blockIdx.x * blockDim.x + threadIdx.x;
  if (idx >= Ktot * Nout) return;
  int r = idx / Nout, col = idx - r * Nout;
  int g = col / Co, co = col - g * Co;
  float v;
  if (r < 5 * Cx) { int k = r / Cx, ci = r - k * Cx; v = Wx[((g * 5 + k) * Cx + ci) * Co + co]; }
  else { int r2 = r - 5 * Cx; int k = r2 / Ch, ci = r2 - k * Ch; v = Wh[((g * 5 + k) * Ch + ci) * Co + co]; }
  WB[idx] = v;
  if (r == 0) cb[col] = bx[col] + bh[col] + bg[col];
}

// ---------------- Chebyshev basis build ----------------
// T0 blocks: X part (Cx ch) at col 0, H part (Ch ch) at col HB
__global__ void k_t0(const float* __restrict__ X0, const float* __restrict__ H0,
                     float* __restrict__ TB, int RS, int Cx, int Ch, int HB) {
  int i = blockIdx.x, c = threadIdx.x;
  if (c < Cx) TB[(size_t)i * RS + c] = X0[(size_t)i * Cx + c];
  else        TB[(size_t)i * RS + HB + (c - Cx)] = H0[(size_t)i * Ch + (c - Cx)];
}

// init T_k block: zero (k==1) or -T_{k-2}
__global__ void k_binit(float* __restrict__ TB, int RS, int Cx, int oCX, int oPPX,
                        int Ch, int oCH, int oPPH, int zero) {
  int i = blockIdx.x, c = threadIdx.x;
  int off, ppoff;
  if (c < Cx) { off = oCX + c;        ppoff = oPPX + c; }
  else        { off = oCH + (c - Cx); ppoff = oPPH + (c - Cx); }
  float v = 0.0f;
  if (!zero) v = -TB[(size_t)i * RS + ppoff];
  TB[(size_t)i * RS + off] = v;
}

// scatter: T_k[dst] += coef * wnorm * T_{k-1}[src]   (coef = -1 or -2)
__global__ void k_scat(const int* __restrict__ src, const int* __restrict__ dst,
                       const float* __restrict__ wn, float* __restrict__ TB,
                       int RS, int Cx, int oPX, int oCX, int Ch, int oPH, int oCH,
                       float coef) {
  int e = blockIdx.x, c = threadIdx.x;
  float w = coef * wn[e];
  int s = src[e], d = dst[e];
  int po, co;
  if (c < Cx) { po = oPX + c;        co = oCX + c; }
  else        { po = oPH + (c - Cx); co = oCH + (c - Cx); }
  atomicAdd(&TB[(size_t)d * RS + co], w * TB[(size_t)s * RS + po]);
}

// ---------------- dense GEMM via V_WMMA_F32_16X16X4_F32 ----------------
// out[M,Nout] = A[M,Kt] @ W[Kt,Nout] + bias[Nout]
// grid: (ceil(mtiles/8), Nout/16), block 256 (8 waves, one 16x16 tile each)
#define MAXK 320
__global__ void k_gemm(const float* __restrict__ A, const float* __restrict__ W,
                       const float* __restrict__ cb, float* __restrict__ out,
                       int Kt, int Nout, int mtiles, int Nn) {
  __shared__ float bp[MAXK * 16];
  int col0 = blockIdx.y * 16;
  for (int i = threadIdx.x; i < Kt * 16; i += blockDim.x) {
    int r = i >> 4, n = i & 15;
    bp[i] = W[r * Nout + col0 + n];
  }
  __syncthreads();

  int wave = threadIdx.x >> 5, lane = threadIdx.x & 31;
  int mtile = blockIdx.x * 8 + wave;
  if (mtile >= mtiles) return;
  int m0 = mtile << 4;
  int ln = lane & 15, hi = lane >> 4;

  float bias = cb[col0 + ln];
  v8f acc = {bias, bias, bias, bias, bias, bias, bias, bias};

  int arowidx = m0 + ln; if (arowidx >= Nn) arowidx = Nn - 1;
  const float* arow = A + (size_t)arowidx * Kt + hi * 2;
  const float* brow = bp + hi * 2 * 16 + ln;

  #pragma unroll 4
  for (int k = 0; k < Kt; k += 4) {
    v2f a; a.x = arow[k];          a.y = arow[k + 1];          // A[M=ln][K=k+2hi, +1]
    v2f b; b.x = brow[k * 16];     b.y = brow[k * 16 + 16];    // B[K=k+2hi, +1][N=ln]
    acc = __builtin_amdgcn_wmma_f32_16x16x4_f32(
        /*neg_a=*/false, a, /*neg_b=*/false, b,
        /*c_mod=*/(short)0, acc, /*reuse_a=*/false, /*reuse_b=*/false);
  }

  #pragma unroll
  for (int r = 0; r < 8; ++r) {
    int m = m0 + r + hi * 8;
    if (m < Nn) out[(size_t)m * Nout + col0 + ln] = acc[r];
  }
}

// ---------------- LSTM gates with peepholes ----------------
// pre[i, 4C] cols: [i | f | c | o] (biases already folded in)
__global__ void k_gates(const float* __restrict__ pre, const float* __restrict__ wc,
                        float* __restrict__ h, float* __restrict__ cs, int C) {
  int i = blockIdx.x, c = threadIdx.x;
  const float* p = pre + (size_t)i * 4 * C;
  float cp = cs[(size_t)i * C + c];
  float I = sigm(p[c]         + wc[c] * cp);
  float F = sigm(p[C + c]     + wc[C + c] * cp);
  float G = tanhf(p[2 * C + c]);
  float cn = F * cp + I * G;
  float O = sigm(p[3 * C + c] + wc[2 * C + c] * cn);
  cs[(size_t)i * C + c] = cn;
  h[(size_t)i * C + c] = O * tanhf(cn);
}

// ---------------- pooling + head ----------------
__global__ void k_pool(const float* __restrict__ h2, const int* __restrict__ batch,
                       float* __restrict__ u, int N, int C) {
  int t = blockIdx.x * blockDim.x + threadIdx.x;
  if (t >= N * C) return;
  int i = t / C, c = t - i * C;
  atomicAdd(&u[batch[i] * C + c], h2[t]);
}

__global__ void k_final(const float* __restrict__ u, const float* __restrict__ lw,
                        const float* __restrict__ lb, float* __restrict__ out,
                        int G, int C) {
  int g = threadIdx.x;
  if (g >= G) return;
  float s = lb[0];
  for (int c = 0; c < C; ++c) s += u[g * C + c] * lw[c];
  out[g] = s;
}

// ---------------- host orchestration ----------------
extern "C" void kernel_launch(void* const* d_in, const int* in_sizes, int n_in,
                              void* d_out, int out_size, void* d_ws, size_t ws_size,
                              hipStream_t stream) {
  const float* x     = (const float*)d_in[0];
  const int*   ei    = (const int*)d_in[1];
  const float* eattr = (const float*)d_in[2];
  const int*   batch = (const int*)d_in[3];
  const float* Wx1 = (const float*)d_in[4];
  const float* bx1 = (const float*)d_in[5];
  const float* Wh1 = (const float*)d_in[6];
  const float* bh1 = (const float*)d_in[7];
  const float* wc1 = (const float*)d_in[8];
  const float* b1  = (const float*)d_in[9];
  const float* Wx2 = (const float*)d_in[10];
  const float* bx2 = (const float*)d_in[11];
  const float* Wh2 = (const float*)d_in[12];
  const float* bh2 = (const float*)d_in[13];
  const float* wc2 = (const float*)d_in[14];
  const float* b2  = (const float*)d_in[15];
  const float* lw  = (const float*)d_in[16];
  const float* lb  = (const float*)d_in[17];

  const int E = in_sizes[2];
  const int N = in_sizes[3];
  const int F = 32, H1 = 32, H2 = 16, G = 64;
  const int T = in_sizes[0] / (N * F);
  const int* src = ei;
  const int* dst = ei + E;

  // workspace carve (floats)
  float* ws = (float*)d_ws;
  float* wnorm = ws; ws += E;
  float* deg   = ws; ws += N;                  // reused in-place as dis
  float* TB1   = ws; ws += (size_t)N * 320;    // [T0..T4(X)|T0..T4(H)] layer1
  float* TB2   = ws; ws += (size_t)N * 240;    // layer2
  float* pre1  = ws; ws += (size_t)N * 128;
  float* pre2  = ws; ws += (size_t)N * 64;
  float* h1 = ws; ws += (size_t)N * H1;        // h1,c1,h2,c2 contiguous -> one zero pass
  float* c1 = ws; ws += (size_t)N * H1;
  float* h2 = ws; ws += (size_t)N * H2;
  float* c2 = ws; ws += (size_t)N * H2;
  float* WB1 = ws; ws += 320 * 128;
  float* cb1 = ws; ws += 128;
  float* WB2 = ws; ws += 240 * 64;
  float* cb2 = ws; ws += 64;
  float* u   = ws; ws += G * H2;

  auto gz = [](int n) { return dim3((unsigned)((n + 255) / 256)); };

  // graph normalization + weight packing (per launch; deterministic)
  k_zero<<<gz(N), 256, 0, stream>>>(deg, N);
  k_zero<<<gz(N * (2 * H1 + 2 * H2)), 256, 0, stream>>>(h1, N * (2 * H1 + 2 * H2));
  k_deg<<<gz(E), 256, 0, stream>>>(src, eattr, deg, E);
  k_dis<<<gz(N), 256, 0, stream>>>(deg, N);
  k_wnorm<<<gz(E), 256, 0, stream>>>(src, dst, eattr, deg, wnorm, E);
  k_pack<<<gz(320 * 128), 256, 0, stream>>>(Wx1, Wh1, bx1, bh1, b1, WB1, cb1, 32, 32, 32);
  k_pack<<<gz(240 * 64), 256, 0, stream>>>(Wx2, Wh2, bx2, bh2, b2, WB2, cb2, 32, 16, 16);

  const int mtiles = (N + 15) / 16;
  dim3 gg1((mtiles + 7) / 8, 128 / 16);
  dim3 gg2((mtiles + 7) / 8, 64 / 16);

  for (int t = 0; t < T; ++t) {
    const float* xt = x + (size_t)t * N * F;

    // ---- layer 1: basis(X=x_t, H=h1), GEMM, gates ----
    k_t0<<<N, 64, 0, stream>>>(xt, h1, TB1, 320, 32, 32, 160);
    for (int k = 1; k <= 4; ++k) {
      int pp = (k >= 2) ? (k - 2) : 0;
      k_binit<<<N, 64, 0, stream>>>(TB1, 320, 32, k * 32, pp * 32,
                                    32, 160 + k * 32, 160 + pp * 32, k == 1);
      k_scat<<<E, 64, 0, stream>>>(src, dst, wnorm, TB1, 320,
                                   32, (k - 1) * 32, k * 32,
                                   32, 160 + (k - 1) * 32, 160 + k * 32,
                                   (k == 1) ? -1.0f : -2.0f);
    }
    k_gemm<<<gg1, 256, 0, stream>>>(TB1, WB1, cb1, pre1, 320, 128, mtiles, N);
    k_gates<<<N, 32, 0, stream>>>(pre1, wc1, h1, c1, 32);

    // ---- layer 2: basis(X=h1, H=h2), GEMM, gates ----
    k_t0<<<N, 48, 0, stream>>>(h1, h2, TB2, 240, 32, 16, 160);
    for (int k = 1; k <= 4; ++k) {
      int pp = (k >= 2) ? (k - 2) : 0;
      k_binit<<<N, 48, 0, stream>>>(TB2, 240, 32, k * 32, pp * 32,
                                    16, 160 + k * 16, 160 + pp * 16, k == 1);
      k_scat<<<E, 48, 0, stream>>>(src, dst, wnorm, TB2, 240,
                                   32, (k - 1) * 32, k * 32,
                                   16, 160 + (k - 1) * 16, 160 + k * 16,
                                   (k == 1) ? -1.0f : -2.0f);
    }
    k_gemm<<<gg2, 256, 0, stream>>>(TB2, WB2, cb2, pre2, 240, 64, mtiles, N);
    k_gates<<<N, 16, 0, stream>>>(pre2, wc2, h2, c2, 16);
  }

  // global add pool + linear head
  k_zero<<<gz(G * H2), 256, 0, stream>>>(u, G * H2);
  k_pool<<<gz(N * H2), 256, 0, stream>>>(h2, batch, u, N, H2);
  k_final<<<1, 64, 0, stream>>>(u, lw, lb, (float*)d_out, G, H2);
}